// Huber_33603824124177
// MI455X (gfx1250) — compile-verified
//
#include <hip/hip_runtime.h>
#include <math.h>

// Problem constants (match reference)
#define MDIM 4096
#define NDIM 4096
#define RDIM 16
#define NLAYERS 3
#define HUB_ITERS 2
#define EPS_RIDGE 1e-5f
#define ROWSPLIT 32          // split reduction dim across 32 blocks -> 512 WGs per pass
#define ROWCHUNK (MDIM / ROWSPLIT)

typedef __attribute__((ext_vector_type(2))) float v2f;
typedef __attribute__((ext_vector_type(4))) float v4f;
typedef __attribute__((ext_vector_type(8))) float v8f;

// ---------------- workspace layout (float offsets) ----------------
enum : size_t {
  OFF_UCUR = 0,
  OFF_VT   = OFF_UCUR + (size_t)MDIM * RDIM,
  OFF_A    = OFF_VT   + (size_t)NDIM * RDIM,
  OFF_T    = OFF_A    + (size_t)4096 * 256,
  OFF_SUMP = OFF_T    + (size_t)4096 * RDIM,
  OFF_SIG  = OFF_SUMP + 4096,
  OFF_CNTC = OFF_SIG  + 4096,
  OFF_CNTR = OFF_CNTC + 4096,
};

// ---------------- small utility kernels ----------------
__global__ void copyKernel(float* __restrict__ dst, const float* __restrict__ src, int n) {
  int i = blockIdx.x * blockDim.x + threadIdx.x;
  if (i < n) dst[i] = src[i];
}

__global__ void zeroKernel(float* __restrict__ dst, int n) {
  int i = blockIdx.x * blockDim.x + threadIdx.x;
  if (i < n) dst[i] = 0.0f;
}

// Vt[n*16+r] = V[r*N + n]
__global__ void vtransKernel(float* __restrict__ Vt, const float* __restrict__ V) {
  int idx = blockIdx.x * blockDim.x + threadIdx.x;   // 65536
  int n = idx >> 4, r = idx & 15;
  Vt[idx] = V[(size_t)r * NDIM + n];
}

// 4096x4096 tiled transpose (XT kept in d_out until the final GEMM overwrites it)
__global__ void transposeKernel(const float* __restrict__ src, float* __restrict__ dst) {
  __shared__ float tile[32][33];
  int bx = blockIdx.x * 32, by = blockIdx.y * 32;
  int tx = threadIdx.x, ty = threadIdx.y;            // block (32,8)
#pragma unroll
  for (int j = 0; j < 32; j += 8)
    tile[ty + j][tx] = src[(size_t)(by + ty + j) * NDIM + bx + tx];
  __syncthreads();
#pragma unroll
  for (int j = 0; j < 32; j += 8)
    dst[(size_t)(bx + ty + j) * MDIM + by + tx] = tile[tx][ty + j];
}

// per-column observed count (split over rows + atomics)
__global__ void countKernel(const float* __restrict__ src, float* __restrict__ cnt) {
  int col = blockIdx.x * blockDim.x + threadIdx.x;
  int i0 = blockIdx.y * ROWCHUNK;
  float a = 0.0f;
  for (int i = i0; i < i0 + ROWCHUNK; ++i)
    a += (src[(size_t)i * NDIM + col] != 0.0f) ? 1.0f : 0.0f;
  atomicAdd(cnt + col, a);
}

__global__ void sigInitKernel(float* __restrict__ sig, const float* __restrict__ sigma) {
  int i = blockIdx.x * blockDim.x + threadIdx.x;
  sig[i] = sigma[0];
}

// sig = lamda * sqrt(sum psi^2) / sqrt(2*cnt*alpha); alpha via closed-form chi2 cdfs
__global__ void sigUpdateKernel(const float* __restrict__ sump, const float* __restrict__ cnt,
                                float* __restrict__ sig, const float* __restrict__ cvec,
                                const float* __restrict__ lvec, int layer) {
  int i = blockIdx.x * blockDim.x + threadIdx.x;
  float cl = cvec[layer], ll = lvec[layer];
  // chi2.cdf(2c,1) = erf(sqrt(c)) ; chi2.cdf(c^2,3) = erf(c/sqrt2) - sqrt(2/pi)*c*exp(-c^2/2)
  float alpha = cl * (1.0f - erff(sqrtf(cl)))
              + 0.5f * (erff(cl * 0.70710678118654752f)
                        - 0.79788456080286536f * cl * expf(-0.5f * cl * cl));
  float tau = sqrtf(sump[i]) / sqrtf(2.0f * cnt[i] * alpha);
  sig[i] = tau * ll;
}

// ---------------- WMMA Gram kernel ----------------
// A[oc][r][s] = sum_k w(src[k][oc]) * F[k][r] * F[k][s] + eps*I
//   V-step: src = X  (k = row m),  F = Ucur, oc = column n
//   U-step: src = XT (k = col n),  F = Vt,   oc = row m
// Block = 8 waves = 4 r-groups x 2 k-halves; each wave owns 4 r-values (4 v8f accs)
// so the w-fragment (A-matrix) is built once per 4 WMMAs, and the 4 per-r scale
// factors are fetched with a single b128 load. K-halves reduced through LDS.
__global__ void __launch_bounds__(256) gramKernel(const float* __restrict__ src,
                                                  const float* __restrict__ F,
                                                  float* __restrict__ A) {
  __shared__ float red[4][32][32];                 // [rgroup][lane][j*8+i] partials, 16KB

  const int oc0  = blockIdx.x * 16;
  const int wv   = threadIdx.x >> 5;
  const int rg   = wv & 3;                         // r-group: r = 4*rg + j
  const int kh   = wv >> 2;                        // k-half: 0 or 1
  const int lane = threadIdx.x & 31;
  const int half = lane >> 4;                      // K split within WMMA frag
  const int col  = lane & 15;

  v8f acc[4];
#pragma unroll
  for (int j = 0; j < 4; ++j)
    acc[j] = (v8f){0.f, 0.f, 0.f, 0.f, 0.f, 0.f, 0.f, 0.f};

  const int kbase = kh * (MDIM / 2) + 2 * half;
  const float* sp = src + (size_t)kbase * NDIM + oc0 + col;   // w source rows
  const float* fc = F + kbase * RDIM + col;                   // F[k][col] (per-lane)
  const float* fr = F + kbase * RDIM + 4 * rg;                // F[k][4rg..4rg+3] (uniform/half)

  for (int k0 = 0; k0 < MDIM / 2; k0 += 4) {
    v2f a;
    a.x = (sp[0]    != 0.0f) ? 1.0f : 0.0f;        // w[k  ][oc0+col]
    a.y = (sp[NDIM] != 0.0f) ? 1.0f : 0.0f;        // w[k+1][oc0+col]
    float fca = fc[0], fcb = fc[RDIM];             // F[k][col], F[k+1][col]
    v4f fra = *(const v4f*)fr;                     // F[k][4rg+j], one b128
    v4f frb = *(const v4f*)(fr + RDIM);            // F[k+1][4rg+j]
#pragma unroll
    for (int j = 0; j < 4; ++j) {
      v2f b;
      b.x = fca * fra[j];
      b.y = fcb * frb[j];
      acc[j] = __builtin_amdgcn_wmma_f32_16x16x4_f32(false, a, false, b,
                                                     (short)0, acc[j], false, false);
    }
    sp += (size_t)4 * NDIM;
    fc += 4 * RDIM;
    fr += 4 * RDIM;
  }

  // reduce the two k-halves through LDS
  if (kh == 1) {
#pragma unroll
    for (int j = 0; j < 4; ++j)
#pragma unroll
      for (int i = 0; i < 8; ++i) red[rg][lane][j * 8 + i] = acc[j][i];
  }
  __syncthreads();
  if (kh == 0) {
#pragma unroll
    for (int j = 0; j < 4; ++j) {
#pragma unroll
      for (int i = 0; i < 8; ++i) acc[j][i] += red[rg][lane][j * 8 + i];
      int r = 4 * rg + j;
      if (col == r) {                              // ridge on diagonal (s == r)
#pragma unroll
        for (int i = 0; i < 8; ++i) acc[j][i] += EPS_RIDGE;
      }
#pragma unroll
      for (int i = 0; i < 8; ++i)                  // D vgpr i: rows oc=i / oc=8+i
        A[(size_t)(oc0 + i + 8 * half) * 256 + r * 16 + col] = acc[j][i];
    }
  }
}

// ---------------- batched 16x16 inverse (one wave per matrix) ----------------
// Gauss-Jordan on augmented [A | I], one column per lane (lanes 0-15 = A, 16-31 = I).
// SPD + ridge -> no pivoting. Inverse written back in place.
__global__ void invertKernel(float* __restrict__ A) {
  int lane = threadIdx.x & 31;
  int mat  = blockIdx.x * 4 + (threadIdx.x >> 5);   // block = 128 thr = 4 waves
  float* Am = A + (size_t)mat * 256;

  float x[16];
#pragma unroll
  for (int i = 0; i < 16; ++i)
    x[i] = (lane < 16) ? Am[i * 16 + lane] : ((i == lane - 16) ? 1.0f : 0.0f);

#pragma unroll
  for (int k = 0; k < 16; ++k) {
    float piv = __shfl(x[k], k);
    float xk  = x[k] * (1.0f / piv);
#pragma unroll
    for (int i = 0; i < 16; ++i) {
      float f = __shfl(x[i], k);              // column-k value of row i (before update)
      if (i != k) x[i] -= f * xk;
    }
    x[k] = xk;
  }
  if (lane >= 16) {
#pragma unroll
    for (int i = 0; i < 16; ++i) Am[i * 16 + (lane - 16)] = x[i];
  }
}

// ---------------- IRLS passes (bandwidth-bound, register-resident R=16) ----------------
__global__ void iterAKernel(const float* __restrict__ src,  // 4096x4096
                            const float* __restrict__ Fx,   // fixed factor per column (4096x16)
                            const float* __restrict__ Bc,   // broadcast factor per row (4096x16)
                            const float* __restrict__ sig,
                            float* __restrict__ sump,
                            const float* __restrict__ cvec, int layer) {
  int col = blockIdx.x * blockDim.x + threadIdx.x;
  int i0 = blockIdx.y * ROWCHUNK;
  float cl = cvec[layer];
  float f[16];
#pragma unroll
  for (int r = 0; r < 16; ++r) f[r] = Fx[(size_t)col * 16 + r];
  float sinv = 1.0f / sig[col];
  float acc = 0.0f;
  for (int i = i0; i < i0 + ROWCHUNK; ++i) {
    float x = src[(size_t)i * NDIM + col];
    float pred = 0.0f;
#pragma unroll
    for (int r = 0; r < 16; ++r) pred += Bc[(size_t)i * 16 + r] * f[r];
    float res = (x != 0.0f) ? (x - pred) : 0.0f;
    float psi = fminf(fmaxf(res * sinv, -cl), cl);
    acc += psi * psi;
  }
  atomicAdd(sump + col, acc);
}

// iterB1: t[col][r] = sum_i Bc[i][r] * (clip(res/sig)*sig)
__global__ void iterB1Kernel(const float* __restrict__ src,
                             const float* __restrict__ Fx,
                             const float* __restrict__ Bc,
                             const float* __restrict__ sig,
                             float* __restrict__ t,
                             const float* __restrict__ cvec, int layer) {
  int col = blockIdx.x * blockDim.x + threadIdx.x;
  int i0 = blockIdx.y * ROWCHUNK;
  float cl = cvec[layer];
  float f[16], acc[16];
#pragma unroll
  for (int r = 0; r < 16; ++r) { f[r] = Fx[(size_t)col * 16 + r]; acc[r] = 0.0f; }
  float s = sig[col], sinv = 1.0f / s;
  for (int i = i0; i < i0 + ROWCHUNK; ++i) {
    float x = src[(size_t)i * NDIM + col];
    float b[16];
    float pred = 0.0f;
#pragma unroll
    for (int r = 0; r < 16; ++r) { b[r] = Bc[(size_t)i * 16 + r]; pred += b[r] * f[r]; }
    float res = (x != 0.0f) ? (x - pred) : 0.0f;
    float psi2 = fminf(fmaxf(res * sinv, -cl), cl);
    float val = psi2 * s;
#pragma unroll
    for (int r = 0; r < 16; ++r) acc[r] += b[r] * val;
  }
#pragma unroll
  for (int r = 0; r < 16; ++r) atomicAdd(t + (size_t)col * 16 + r, acc[r]);
}

// iterB2: beta += mu * (Ainv @ t), one thread per (col, r)
__global__ void deltaKernel(float* __restrict__ Fx, const float* __restrict__ t,
                            const float* __restrict__ Ainv,
                            const float* __restrict__ mvec, int layer) {
  int idx = blockIdx.x * blockDim.x + threadIdx.x;   // 65536
  int col = idx >> 4, r = idx & 15;
  float ml = mvec[layer];
  const float* Am = Ainv + (size_t)col * 256 + r * 16;
  const float* tv = t + (size_t)col * 16;
  float d = 0.0f;
#pragma unroll
  for (int s = 0; s < 16; ++s) d += Am[s] * tv[s];
  Fx[idx] += ml * d;
}

// out = Ucur @ V  (writes d_out, overwriting XT scratch)
__global__ void finalKernel(const float* __restrict__ Ucur, const float* __restrict__ Vt,
                            float* __restrict__ out) {
  int n = blockIdx.x * blockDim.x + threadIdx.x;
  int m0 = blockIdx.y * ROWCHUNK;
  float f[16];
#pragma unroll
  for (int r = 0; r < 16; ++r) f[r] = Vt[(size_t)n * 16 + r];
  for (int m = m0; m < m0 + ROWCHUNK; ++m) {
    float pred = 0.0f;
#pragma unroll
    for (int r = 0; r < 16; ++r) pred += Ucur[(size_t)m * 16 + r] * f[r];
    out[(size_t)m * NDIM + n] = pred;
  }
}

// ---------------- host-side orchestration ----------------
extern "C" void kernel_launch(void* const* d_in, const int* in_sizes, int n_in,
                              void* d_out, int out_size, void* d_ws, size_t ws_size,
                              hipStream_t stream) {
  const float* U     = (const float*)d_in[0];
  const float* V     = (const float*)d_in[1];
  const float* X     = (const float*)d_in[2];
  const float* c     = (const float*)d_in[3];
  const float* lam   = (const float*)d_in[4];
  const float* mu    = (const float*)d_in[5];
  const float* sigma = (const float*)d_in[6];

  float* ws   = (float*)d_ws;
  float* Ucur = ws + OFF_UCUR;
  float* Vt   = ws + OFF_VT;
  float* A    = ws + OFF_A;
  float* t    = ws + OFF_T;
  float* sump = ws + OFF_SUMP;
  float* sig  = ws + OFF_SIG;
  float* cntc = ws + OFF_CNTC;
  float* cntr = ws + OFF_CNTR;
  float* XT   = (float*)d_out;   // 64MB transpose scratch; overwritten by final GEMM
  float* out  = (float*)d_out;

  const dim3 B256(256);
  const dim3 colGrid(NDIM / 256, ROWSPLIT);   // (16, 32) = 512 WGs per pass
  const dim3 g16(NDIM / 256);                 // 4096 threads

  // init: copy factors, build X^T, observed counts
  copyKernel<<<dim3(256), B256, 0, stream>>>(Ucur, U, MDIM * RDIM);
  vtransKernel<<<dim3(256), B256, 0, stream>>>(Vt, V);
  transposeKernel<<<dim3(128, 128), dim3(32, 8), 0, stream>>>(X, XT);
  zeroKernel<<<dim3(32), B256, 0, stream>>>(cntc, 2 * NDIM);  // cntc+cntr contiguous
  countKernel<<<colGrid, B256, 0, stream>>>(X, cntc);
  countKernel<<<colGrid, B256, 0, stream>>>(XT, cntr);

  for (int layer = 0; layer < NLAYERS; ++layer) {
    // ---------- V step (per-column, src = X, fixed = Vt, broadcast = Ucur) ----------
    gramKernel<<<dim3(NDIM / 16), dim3(256), 0, stream>>>(X, Ucur, A);
    invertKernel<<<dim3(NDIM / 4), dim3(128), 0, stream>>>(A);
    sigInitKernel<<<g16, B256, 0, stream>>>(sig, sigma);
    for (int it = 0; it < HUB_ITERS; ++it) {
      zeroKernel<<<dim3((NDIM * RDIM + NDIM + 255) / 256), B256, 0, stream>>>(t, NDIM * RDIM + NDIM);
      iterAKernel<<<colGrid, B256, 0, stream>>>(X, Vt, Ucur, sig, sump, c, layer);
      sigUpdateKernel<<<g16, B256, 0, stream>>>(sump, cntc, sig, c, lam, layer);
      iterB1Kernel<<<colGrid, B256, 0, stream>>>(X, Vt, Ucur, sig, t, c, layer);
      deltaKernel<<<dim3(NDIM * RDIM / 256), B256, 0, stream>>>(Vt, t, A, mu, layer);
    }
    // ---------- U step (per-row, src = XT, fixed = Ucur, broadcast = Vt) ----------
    gramKernel<<<dim3(MDIM / 16), dim3(256), 0, stream>>>(XT, Vt, A);
    invertKernel<<<dim3(MDIM / 4), dim3(128), 0, stream>>>(A);
    sigInitKernel<<<g16, B256, 0, stream>>>(sig, sigma);
    for (int it = 0; it < HUB_ITERS; ++it) {
      zeroKernel<<<dim3((MDIM * RDIM + MDIM + 255) / 256), B256, 0, stream>>>(t, MDIM * RDIM + MDIM);
      iterAKernel<<<colGrid, B256, 0, stream>>>(XT, Ucur, Vt, sig, sump, c, layer);
      sigUpdateKernel<<<g16, B256, 0, stream>>>(sump, cntr, sig, c, lam, layer);
      iterB1Kernel<<<colGrid, B256, 0, stream>>>(XT, Ucur, Vt, sig, t, c, layer);
      deltaKernel<<<dim3(MDIM * RDIM / 256), B256, 0, stream>>>(Ucur, t, A, mu, layer);
    }
  }

  // out = Ucur @ V (overwrites XT scratch)
  finalKernel<<<colGrid, B256, 0, stream>>>(Ucur, Vt, out);
}